// SymmetricRandomFeatures_6957847019755
// MI455X (gfx1250) — compile-verified
//
#include <hip/hip_runtime.h>
#include <math.h>

typedef __attribute__((ext_vector_type(2))) float v2f;
typedef __attribute__((ext_vector_type(8))) float v8f;
typedef __attribute__((ext_vector_type(4))) int   v4i;

#define SEQ  262144
#define DK   64
#define LSTR 72   // LDS row stride in floats: 288B (16B aligned), rows+2 -> +16 banks

// workspace layout (floats)
#define WS_GQ   0        // 64x64 gram(q) * 1/S
#define WS_GK   4096     // 64x64 gram(k) * 1/S
#define WS_SQ   8192     // 64 column sums of q
#define WS_SK   8256     // 64 column sums of k
#define WS_B    8320     // 64x64 B^T  (Bt[d][m] = B[m][d])
#define WS_BIAS 12416    // 64 feat_bias
#define WS_ZERO_COUNT 8320

#if defined(__gfx1250__) && __has_builtin(__builtin_amdgcn_global_load_async_to_lds_b128)
#define HAVE_ASYNC 1
#if __has_builtin(__builtin_amdgcn_s_wait_asynccnt)
#define ASYNC_WAIT(n) __builtin_amdgcn_s_wait_asynccnt(n)
#else
#define ASYNC_WAIT(n) asm volatile("s_wait_asynccnt %0" ::"n"(n) : "memory")
#endif
// probe-derived signature: (int4* global_src, int4* lds_dst, imm offset, imm cpol)
#define ASYNC_CP16(gp, lp) \
  __builtin_amdgcn_global_load_async_to_lds_b128((v4i*)(gp), (v4i*)(lp), 0, 0)
#else
#define HAVE_ASYNC 0
#endif

__global__ void k_zero(float* __restrict__ ws) {
  for (int i = blockIdx.x * blockDim.x + threadIdx.x; i < WS_ZERO_COUNT;
       i += gridDim.x * blockDim.x)
    ws[i] = 0.0f;
}

// --- Phase 1: Gram matrices + column sums -----------------------------------
__global__ __launch_bounds__(512) void k_gram(const float* __restrict__ q,
                                              const float* __restrict__ k,
                                              float* __restrict__ ws) {
  __shared__ float tile[2][32 * LSTR];   // double-buffered 32x64 row chunk
  __shared__ float csum[8 * 64];

  const float* src = blockIdx.y ? k : q;
  float* G    = ws + (blockIdx.y ? WS_GK : WS_GQ);
  float* Svec = ws + (blockIdx.y ? WS_SK : WS_SQ);

  const int tid  = threadIdx.x;
  const int lane = tid & 31;            // wave32
  const int wid  = tid >> 5;            // 16 waves
  const int mi   = wid >> 2;            // 4x4 tile grid over 64x64 output
  const int nj   = wid & 3;
  const long row0 = (long)blockIdx.x * 1024;

  const int aL = lane & 15;             // M / N index within tile
  const int kb = (lane >> 4) * 2;       // K pair base (0 or 2)
  const int sc = tid & 63;              // column for mean sums
  const int sr = tid >> 6;              // 0..7 row group
  const int ldr = tid >> 4;             // staging row 0..31
  const int ldc = (tid & 15) * 4;       // staging col

  v8f acc = {0.f,0.f,0.f,0.f,0.f,0.f,0.f,0.f};
  float colsum = 0.0f;

#if HAVE_ASYNC
  // prime buffer 0
  ASYNC_CP16(&src[(row0 + ldr) * 64 + ldc], &tile[0][ldr * LSTR + ldc]);
#endif

  for (int ch = 0; ch < 32; ++ch) {
    const int cur = ch & 1;
#if HAVE_ASYNC
    if (ch < 31) {
      ASYNC_CP16(&src[(row0 + (ch + 1) * 32 + ldr) * 64 + ldc],
                 &tile[1 - cur][ldr * LSTR + ldc]);
      ASYNC_WAIT(1);   // chunk ch (issued earlier) complete for this wave
    } else {
      ASYNC_WAIT(0);
    }
#else
    const float4 v =
        *(const float4*)&src[(row0 + ch * 32 + ldr) * 64 + ldc];
    float* d = &tile[cur][ldr * LSTR + ldc];
    d[0] = v.x; d[1] = v.y; d[2] = v.z; d[3] = v.w;
#endif
    __syncthreads();
    const float* tl = tile[cur];

    // partial column sums (for means)
    colsum += tl[(sr     ) * LSTR + sc] + tl[(sr +  8) * LSTR + sc]
            + tl[(sr + 16) * LSTR + sc] + tl[(sr + 24) * LSTR + sc];

    // C[m,n] += sum_s q[s,m]*q[s,n] ; A = q^T frag, B = q frag
#pragma unroll
    for (int s = 0; s < 32; s += 4) {
      v2f a, b;
      a.x = tl[(s + kb    ) * LSTR + mi * 16 + aL];
      a.y = tl[(s + kb + 1) * LSTR + mi * 16 + aL];
      b.x = tl[(s + kb    ) * LSTR + nj * 16 + aL];
      b.y = tl[(s + kb + 1) * LSTR + nj * 16 + aL];
      acc = __builtin_amdgcn_wmma_f32_16x16x4_f32(false, a, false, b,
                                                  (short)0, acc, false, false);
    }
    __syncthreads();
  }

  // reduce column sums across the 8 row-groups, merge globally
  csum[sr * 64 + sc] = colsum;
  __syncthreads();
  if (tid < 64) {
    float t = 0.0f;
#pragma unroll
    for (int r = 0; r < 8; ++r) t += csum[r * 64 + tid];
    atomicAdd(&Svec[tid], t);
  }

  // merge gram tile (scale by 1/S here)
  const float inv_s = 1.0f / (float)SEQ;
  const int moff = (lane >> 4) * 8;
#pragma unroll
  for (int r = 0; r < 8; ++r) {
    const int m = mi * 16 + r + moff;
    const int n = nj * 16 + aL;
    atomicAdd(&G[m * 64 + n], acc[r] * inv_s);
  }
}

// --- Phase 2: 64x64 symmetric eigensolve (parallel cyclic Jacobi) -----------
__global__ __launch_bounds__(64) void k_eig(const float* __restrict__ rm,
                                            const float* __restrict__ noise,
                                            float* __restrict__ ws) {
  __shared__ float A[64 * 65];
  __shared__ float V[64 * 65];
  __shared__ float mu4[64], mu5[64];
  __shared__ int   perm[64];
  __shared__ int   pA[32], qA[32];
  __shared__ float cA[32], sA[32];
  __shared__ float lam[64], red[64];
  __shared__ float Dsh;

  const int t = threadIdx.x;
  const float inv_s = 1.0f / (float)SEQ;

  mu4[t] = ws[WS_SQ + t] * inv_s;
  mu5[t] = ws[WS_SK + t] * inv_s;
  perm[t] = t;
  __syncthreads();

  for (int j = 0; j < 64; ++j) {
    A[t * 65 + j] = ws[WS_GQ + t * 64 + j] + ws[WS_GK + t * 64 + j]
                  + mu4[t] * mu5[j] + mu5[t] * mu4[j];
    V[t * 65 + j] = (t == j) ? 1.0f : 0.0f;
  }
  __syncthreads();

  for (int sweep = 0; sweep < 12; ++sweep) {
    for (int round = 0; round < 63; ++round) {
      if (t < 32) {
        int p = perm[t], qi = perm[63 - t];
        if (p > qi) { int tmp = p; p = qi; qi = tmp; }
        const float app = A[p * 65 + p];
        const float aqq = A[qi * 65 + qi];
        const float apq = A[p * 65 + qi];
        float c = 1.0f, s = 0.0f;
        if (fabsf(apq) > 1e-12f) {
          const float tau = (aqq - app) / (2.0f * apq);
          const float tt  = (tau >= 0.0f ? 1.0f : -1.0f)
                            / (fabsf(tau) + sqrtf(1.0f + tau * tau));
          c = rsqrtf(1.0f + tt * tt);
          s = tt * c;
        }
        pA[t] = p; qA[t] = qi; cA[t] = c; sA[t] = s;
      }
      __syncthreads();
      // A <- A*J ; V <- V*J (thread = row; 32 disjoint column pairs)
      for (int j = 0; j < 32; ++j) {
        const int p = pA[j], qi = qA[j];
        const float c = cA[j], s = sA[j];
        float x = A[t * 65 + p], y = A[t * 65 + qi];
        A[t * 65 + p]  = c * x - s * y;
        A[t * 65 + qi] = s * x + c * y;
        x = V[t * 65 + p]; y = V[t * 65 + qi];
        V[t * 65 + p]  = c * x - s * y;
        V[t * 65 + qi] = s * x + c * y;
      }
      __syncthreads();
      // A <- J^T*A (thread = column; 32 disjoint row pairs)
      for (int j = 0; j < 32; ++j) {
        const int p = pA[j], qi = qA[j];
        const float c = cA[j], s = sA[j];
        const float x = A[p * 65 + t], y = A[qi * 65 + t];
        A[p * 65 + t]  = c * x - s * y;
        A[qi * 65 + t] = s * x + c * y;
      }
      __syncthreads();
      if (t == 0) {  // round-robin rotation (circle method), slot 0 fixed
        const int last = perm[63];
        for (int i = 63; i > 1; --i) perm[i] = perm[i - 1];
        perm[1] = last;
      }
      __syncthreads();
    }
  }

  lam[t] = A[t * 65 + t];
  __syncthreads();
  // descending rank (mirror SVD ordering), tie-broken by index
  int rank = 0;
  for (int j = 0; j < 64; ++j)
    rank += (lam[j] > lam[t]) || (lam[j] == lam[t] && j < t);

  const float l   = lam[t];
  const float a   = (1.0f - 2.0f * l
                     - sqrtf((2.0f * l + 1.0f) * (2.0f * l + 1.0f) + 8.0f * l))
                    / 16.0f;
  const float om4 = 1.0f - 4.0f * a;  // >= 1
  red[t] = logf(om4);
  __syncthreads();
  if (t == 0) {
    float sum = 0.0f;
    for (int j = 0; j < 64; ++j) sum += red[j];
    Dsh = expf(0.25f * sum);
  }
  __syncthreads();

  // store B transposed: Bt[j][rank] = sqrt(om4) * V[j][t]
  const float sq = sqrtf(om4);
  for (int j = 0; j < 64; ++j)
    ws[WS_B + j * 64 + rank] = sq * V[j * 65 + t];

  // feat_bias[t] = 0.5 * sum_i (RM @ noise^T)[i,t]^2 + D
  float accb = 0.0f;
  for (int i = 0; i < 64; ++i) {
    float w = 0.0f;
    for (int d = 0; d < 64; ++d) w += rm[i * 64 + d] * noise[t * 64 + d];
    accb += w * w;
  }
  ws[WS_BIAS + t] = 0.5f * accb + Dsh;
}

// --- Phase 3: out = exp(x @ B^T + bias) -------------------------------------
__global__ __launch_bounds__(512) void k_apply(const float* __restrict__ q,
                                               const float* __restrict__ k,
                                               const float* __restrict__ ws,
                                               float* __restrict__ out) {
  __shared__ float Bs[64 * LSTR];    // Bt[d][m], staged 1:1 from ws
  __shared__ float xs[256 * LSTR];   // this block's 256 input rows
  __shared__ float bias[64];

  const float* src = blockIdx.y ? k : q;
  float* dst = out + (size_t)blockIdx.y * (size_t)SEQ * 64;

  const int tid  = threadIdx.x;
  const int lane = tid & 31;
  const int wid  = tid >> 5;
  const long rowblk = (long)blockIdx.x * 256;

#if HAVE_ASYNC
  {
    const int r = tid >> 4, c4 = (tid & 15) * 4;       // Bt: 1024 float4
    ASYNC_CP16(&ws[WS_B + r * 64 + c4], &Bs[r * LSTR + c4]);
    ASYNC_CP16(&ws[WS_B + (r + 32) * 64 + c4], &Bs[(r + 32) * LSTR + c4]);
  }
  for (int i = tid; i < 4096; i += 512) {              // x: 4096 float4
    const int r = i >> 4, c4 = (i & 15) * 4;
    ASYNC_CP16(&src[(rowblk + r) * 64 + c4], &xs[r * LSTR + c4]);
  }
  ASYNC_WAIT(0);
#else
  {
    const int r = tid >> 4, c4 = (tid & 15) * 4;
    *(float4*)&Bs[r * LSTR + c4] = *(const float4*)&ws[WS_B + r * 64 + c4];
    *(float4*)&Bs[(r + 32) * LSTR + c4] =
        *(const float4*)&ws[WS_B + (r + 32) * 64 + c4];
  }
  for (int i = tid; i < 4096; i += 512) {
    const int r = i >> 4, c4 = (i & 15) * 4;
    *(float4*)&xs[r * LSTR + c4] = *(const float4*)&src[(rowblk + r) * 64 + c4];
  }
#endif
  if (tid < 64) bias[tid] = ws[WS_BIAS + tid];
  __syncthreads();

  const int aL = lane & 15;
  const int kb = (lane >> 4) * 2;
  const int arow = wid * 16 + aL;   // this lane's input row within the block

  v8f acc0 = {0.f,0.f,0.f,0.f,0.f,0.f,0.f,0.f};
  v8f acc1 = acc0, acc2 = acc0, acc3 = acc0;

#pragma unroll
  for (int d = 0; d < 64; d += 4) {
    v2f a;   // adjacent pair -> single ds_load_b64
    a.x = xs[arow * LSTR + d + kb];
    a.y = xs[arow * LSTR + d + kb + 1];
    v2f b0, b1, b2, b3;
    b0.x = Bs[(d + kb) * LSTR +  0 + aL]; b0.y = Bs[(d + kb + 1) * LSTR +  0 + aL];
    b1.x = Bs[(d + kb) * LSTR + 16 + aL]; b1.y = Bs[(d + kb + 1) * LSTR + 16 + aL];
    b2.x = Bs[(d + kb) * LSTR + 32 + aL]; b2.y = Bs[(d + kb + 1) * LSTR + 32 + aL];
    b3.x = Bs[(d + kb) * LSTR + 48 + aL]; b3.y = Bs[(d + kb + 1) * LSTR + 48 + aL];
    acc0 = __builtin_amdgcn_wmma_f32_16x16x4_f32(false, a, false, b0, (short)0, acc0, false, false);
    acc1 = __builtin_amdgcn_wmma_f32_16x16x4_f32(false, a, false, b1, (short)0, acc1, false, false);
    acc2 = __builtin_amdgcn_wmma_f32_16x16x4_f32(false, a, false, b2, (short)0, acc2, false, false);
    acc3 = __builtin_amdgcn_wmma_f32_16x16x4_f32(false, a, false, b3, (short)0, acc3, false, false);
  }

  const int moff = (lane >> 4) * 8;
#pragma unroll
  for (int r = 0; r < 8; ++r) {
    const long row = rowblk + wid * 16 + r + moff;
    __builtin_nontemporal_store(__expf(acc0[r] + bias[ 0 + aL]), &dst[row * 64 +  0 + aL]);
    __builtin_nontemporal_store(__expf(acc1[r] + bias[16 + aL]), &dst[row * 64 + 16 + aL]);
    __builtin_nontemporal_store(__expf(acc2[r] + bias[32 + aL]), &dst[row * 64 + 32 + aL]);
    __builtin_nontemporal_store(__expf(acc3[r] + bias[48 + aL]), &dst[row * 64 + 48 + aL]);
  }
}

extern "C" void kernel_launch(void* const* d_in, const int* in_sizes, int n_in,
                              void* d_out, int out_size, void* d_ws, size_t ws_size,
                              hipStream_t stream) {
  const float* q     = (const float*)d_in[0];
  const float* k     = (const float*)d_in[1];
  const float* rm    = (const float*)d_in[2];
  const float* noise = (const float*)d_in[3];
  float* out = (float*)d_out;
  float* ws  = (float*)d_ws;

  k_zero<<<32, 256, 0, stream>>>(ws);

  dim3 g1(SEQ / 1024, 2);           // 256 blocks x {q,k}
  k_gram<<<g1, 512, 0, stream>>>(q, k, ws);

  k_eig<<<1, 64, 0, stream>>>(rm, noise, ws);

  dim3 g3(SEQ / 256, 2);            // 1024 blocks x {q,k}
  k_apply<<<g3, 512, 0, stream>>>(q, k, ws, out);
}